// RungeKuttaIntegratorCell_15599321219533
// MI455X (gfx1250) — compile-verified
//
#include <hip/hip_runtime.h>
#include <cstdint>

// RK4 integrator cell, elementwise over 4,194,304 elements.
// Memory-bound (~20 B/elem vs ~25 FLOP/elem): optimized as a streaming
// kernel using CDNA5 async global->LDS copies (ASYNCcnt) for the state
// tile, conflict-free LDS reads of the live column (vel), coalesced
// global loads/stores, and global_prefetch_b8 on the input stream.

#define RK_DT     0.005f
#define RK_INV_M  (1.0f / 95.45f)
#define RK_OFFST  3.2902f
#define RK_F_V    214.9261f
#define RK_F_C    19.3607f
#define RK_DT6    (0.005f / 6.0f)

#define TPB   256                 // 8 wave32 per block
#define EPT   4                   // elements per thread
#define TILE  (TPB * EPT)         // 1024 elements per block
#define TILE_BYTES (TILE * 2 * 4) // 8 KB of states per block

__device__ __forceinline__ float rk_sign(float v) {
    // jnp.sign semantics for finite v: -1 / 0 / +1, branchless.
    return (float)((v > 0.0f) - (v < 0.0f));
}

__device__ __forceinline__ float rk_accel(float u, float vel) {
    return (u + RK_OFFST - RK_F_V * vel - RK_F_C * rk_sign(vel)) * RK_INV_M;
}

__global__ __launch_bounds__(TPB) void rk4_cell_kernel(
    const float* __restrict__ inputs,   // [n]
    const float* __restrict__ states,   // [n][2] (only [:,1] is live)
    float*       __restrict__ out,      // [n][2]
    int n)
{
    __shared__ float s_states[TILE * 2]; // 8 KB staged tile of states

    const int tid  = threadIdx.x;
    const int base = blockIdx.x * TILE;

    // ---- Stage the 8 KB states tile into LDS with CDNA5 async copies ----
    // Each lane issues two 16-byte async loads: lane-contiguous 512 B per
    // wave, ASYNCcnt-tracked, no destination VGPRs consumed.
    {
        const uint64_t gbase = (uint64_t)(uintptr_t)(states + (size_t)base * 2);
        const uint32_t lbase = (uint32_t)(uintptr_t)s_states; // LDS byte addr
        #pragma unroll
        for (int i = 0; i < 2; ++i) {
            const int      boff = tid * 16 + i * (TPB * 16); // 0 .. 8176
            // first element covered by this 16B chunk
            const int      e0   = base + (boff >> 3);
            if (e0 < n) {
                const uint64_t ga = gbase + (uint64_t)boff;
                const uint32_t la = lbase + (uint32_t)boff;
                asm volatile("global_load_async_to_lds_b128 %0, %1, off"
                             :: "v"(la), "v"(ga)
                             : "memory");
            }
        }
    }

    // Prefetch the input tile into cache while the async copies fly
    // (emits CDNA5 global_prefetch_b8; no counters, purely a hint).
    if (base + tid < n) {
        __builtin_prefetch(inputs + base + tid, 0, 0);
    }

    // Wait for this wave's async copies to land in LDS, then make the
    // block-wide tile visible to all waves.
    asm volatile("s_wait_asynccnt 0x0" ::: "memory");
    __syncthreads();

    // ---- Compute: 4 elements per lane, stride-TPB so the LDS vel reads
    //      are 8-byte strided across lanes (conflict-free on 64 banks). ----
    #pragma unroll
    for (int k = 0; k < EPT; ++k) {
        const int el = k * TPB + tid;   // element within tile
        const int e  = base + el;       // global element
        if (e < n) {
            const float u  = inputs[e];
            const float v1 = s_states[2 * el + 1];

            const float a1 = rk_accel(u, v1);
            const float v2 = fmaf(0.5f * RK_DT, a1, v1);
            const float a2 = rk_accel(u, v2);
            const float v4 = fmaf(RK_DT, a1, v1);
            const float a4 = rk_accel(u, v4);

            // st_n = DT/6 * (k1 + 2*k2 + 2*k3 + k4), and k3 == k2 exactly
            const float o0 = RK_DT6 * (v1 + 4.0f * v2 + v4);
            const float o1 = RK_DT6 * (a1 + 4.0f * a2 + a4);

            reinterpret_cast<float2*>(out)[e] = make_float2(o0, o1);
        }
    }
}

extern "C" void kernel_launch(void* const* d_in, const int* in_sizes, int n_in,
                              void* d_out, int out_size, void* d_ws, size_t ws_size,
                              hipStream_t stream) {
    (void)n_in; (void)out_size; (void)d_ws; (void)ws_size;
    const float* inputs = (const float*)d_in[0];   // [n] f32
    const float* states = (const float*)d_in[1];   // [n][2] f32
    float*       out    = (float*)d_out;           // [n][2] f32

    const int n      = in_sizes[0];
    const int blocks = (n + TILE - 1) / TILE;

    hipLaunchKernelGGL(rk4_cell_kernel, dim3(blocks), dim3(TPB), 0, stream,
                       inputs, states, out, n);
}